// LinRegModel_18253611008397
// MI455X (gfx1250) — compile-verified
//
#include <hip/hip_runtime.h>
#include <math.h>

// CDNA5 / gfx1250: wave32, WMMA 16x16x4 f32.
typedef float v2f __attribute__((ext_vector_type(2)));
typedef float v8f __attribute__((ext_vector_type(8)));

#define SEQ 180
#define DIM 256
#define ROWS_PER_BLOCK 16          // one wave per row, 16 waves = 512 threads
#define LDS_STRIDE 260             // 256 + 4 pad -> conflict-free column reads
#define EPS 1e-12f

__launch_bounds__(512)
__global__ void embbag_norm_dot_sigmoid(const int* __restrict__ samples,
                                        const float* __restrict__ emb,
                                        const float* __restrict__ W,
                                        const float* __restrict__ bias,
                                        float* __restrict__ out,
                                        int batch) {
  __shared__ __align__(16) float S[ROWS_PER_BLOCK * LDS_STRIDE];

  const int tid  = threadIdx.x;
  const int lane = tid & 31;
  const int wv   = __builtin_amdgcn_readfirstlane(tid >> 5);  // wave id 0..15 (SGPR)
  const int rowBase = blockIdx.x * ROWS_PER_BLOCK;
  const int row = rowBase + wv;

  // Each lane accumulates dims [lane*4, lane*4+4) and [128 + lane*4, ...):
  // both b128 gathers are fully contiguous 512B across the wave.
  float4 acc0 = make_float4(0.f, 0.f, 0.f, 0.f);
  float4 acc1 = make_float4(0.f, 0.f, 0.f, 0.f);

  if (row < batch) {
    const int* sp = samples + (size_t)row * SEQ;  // wave-uniform -> scalar loads
#pragma unroll 2
    for (int s = 0; s < SEQ; s += 4) {
      int idx[4];
      idx[0] = sp[s + 0];
      idx[1] = sp[s + 1];
      idx[2] = sp[s + 2];
      idx[3] = sp[s + 3];
#pragma unroll
      for (int t = 0; t < 4; ++t) {
        const int   id = idx[t];
        const float m  = (id >= 0) ? 1.0f : 0.0f;         // branch-free masking
        const size_t base = (size_t)((id >= 0) ? id : 0) * DIM;
        const float4* e = (const float4*)(emb + base);
        float4 a = e[lane];        // dims [lane*4 .. +4)
        float4 b = e[lane + 32];   // dims [128 + lane*4 .. +4)
        acc0.x = fmaf(m, a.x, acc0.x);
        acc0.y = fmaf(m, a.y, acc0.y);
        acc0.z = fmaf(m, a.z, acc0.z);
        acc0.w = fmaf(m, a.w, acc0.w);
        acc1.x = fmaf(m, b.x, acc1.x);
        acc1.y = fmaf(m, b.y, acc1.y);
        acc1.z = fmaf(m, b.z, acc1.z);
        acc1.w = fmaf(m, b.w, acc1.w);
      }
    }
  }

  // Stage the 16x256 row-sum tile S into LDS (b128 stores, padded stride).
  float* srow = &S[wv * LDS_STRIDE];
  *(float4*)(srow + lane * 4)       = acc0;
  *(float4*)(srow + 128 + lane * 4) = acc1;
  __syncthreads();

  // ---- Epilogue on wave 0 (EXEC all-ones as WMMA requires) ----
  // A (16x4 f32): lanes 0-15 -> M=lane, K={0,1}; lanes 16-31 -> M=lane-16, K={2,3}
  // B (4x16 f32): VGPR0 lanes 0-15 -> K=0 N=lane, lanes 16-31 -> K=2 N=lane-16; VGPR1 -> K=1/K=3
  // => for B = S^T the B registers equal the A registers: Gram chain is wmma(a, a, c).
  if (tid < 32) {
    const int  rsel = (lane < 16) ? lane : (lane - 16);
    const int  koff = (lane < 16) ? 0 : 2;
    const float* sr = &S[rsel * LDS_STRIDE];

    v8f cg = {0.f, 0.f, 0.f, 0.f, 0.f, 0.f, 0.f, 0.f};  // G = S * S^T
    v8f cp = {0.f, 0.f, 0.f, 0.f, 0.f, 0.f, 0.f, 0.f};  // P = S * Wrep (every col = s.W)

#pragma unroll 4
    for (int kc = 0; kc < DIM; kc += 4) {
      v2f a, wb;
      a.x  = sr[kc + koff + 0];
      a.y  = sr[kc + koff + 1];
      wb.x = W[kc + koff + 0];   // W replicated across all 16 B columns
      wb.y = W[kc + koff + 1];
      cg = __builtin_amdgcn_wmma_f32_16x16x4_f32(false, a, false, a,
                                                 (short)0, cg, false, false);
      cp = __builtin_amdgcn_wmma_f32_16x16x4_f32(false, a, false, wb,
                                                 (short)0, cp, false, false);
    }

    const float bb = bias[0];
    // D element (M,N): VGPR M%8, lane N + 16*(M/8). Diagonal lane also holds P[m][m]=s.W.
#pragma unroll
    for (int i = 0; i < 8; ++i) {
      const int r = (lane < 16) ? i : (i + 8);
      const int c = (lane < 16) ? lane : (lane - 16);
      if (c == r && (rowBase + r) < batch) {
        const float ssq   = cg[i];
        const float dot   = cp[i];
        const float denom = fmaxf(sqrtf(ssq), EPS);
        const float logit = dot / denom + bb;
        out[rowBase + r] = 1.0f / (1.0f + expf(-logit));
      }
    }
  }
}

extern "C" void kernel_launch(void* const* d_in, const int* in_sizes, int n_in,
                              void* d_out, int out_size, void* d_ws, size_t ws_size,
                              hipStream_t stream) {
  (void)n_in; (void)d_ws; (void)ws_size;
  const int*   samples = (const int*)d_in[0];   // [B, 180] token ids, -1 pad
  const float* emb     = (const float*)d_in[1]; // [100000, 256]
  const float* W       = (const float*)d_in[2]; // [1, 256]
  const float* b       = (const float*)d_in[3]; // [1]
  float*       out     = (float*)d_out;         // [B]

  const int batch = in_sizes[0] / SEQ;
  const int grid  = (batch + ROWS_PER_BLOCK - 1) / ROWS_PER_BLOCK;
  embbag_norm_dot_sigmoid<<<grid, 512, 0, stream>>>(samples, emb, W, b, out, batch);
}